// Plastic_SimpleModulated_RNN_60163901882587
// MI455X (gfx1250) — compile-verified
//
#include <hip/hip_runtime.h>

typedef __attribute__((ext_vector_type(2))) float v2f;
typedef __attribute__((ext_vector_type(4))) float v4f;
typedef __attribute__((ext_vector_type(8))) float v8f;

#define Bsz 64
#define Hd  1024
#define Id  512
#define Od  256

// ---------------------------------------------------------------------------
// Kernel 1: i2h preactivation  pre[64,1024] = inputs[64,512] @ Wi2h^T + bi2h
// fp32 WMMA 16x16x4, one wave per 16x16 tile. 256 tiles, K-loop of 128 steps.
// ---------------------------------------------------------------------------
__global__ __launch_bounds__(256) void k_i2h(const float* __restrict__ inp,
                                             const float* __restrict__ Wi2h,
                                             const float* __restrict__ bi2h,
                                             float* __restrict__ pre) {
  const int warp = threadIdx.x >> 5;
  const int lane = threadIdx.x & 31;
  const int t    = blockIdx.x * 8 + warp;       // tile id in [0,256)
  const int m0   = (t >> 6) << 4;               // batch tile (64/16 = 4)
  const int n0   = (t & 63) << 4;               // hidden tile (1024/16 = 64)
  const int l16  = lane & 15;
  const int half = lane >> 4;                   // 0: K={0,1}, 1: K={2,3}
  const float* arow = inp  + (m0 + l16) * Id + (half << 1); // A: inputs row m
  const float* brow = Wi2h + (n0 + l16) * Id + (half << 1); // B: Wi2h^T col n
  v8f acc = {};
  for (int k = 0; k < Id; k += 4) {
    v2f a = *(const v2f*)(arow + k);
    v2f b = *(const v2f*)(brow + k);
    acc = __builtin_amdgcn_wmma_f32_16x16x4_f32(false, a, false, b,
                                                (short)0, acc, false, false);
  }
  const int Mb = m0 + (half << 3);
  const int N  = n0 + l16;
  const float bias = bi2h[N];
#pragma unroll
  for (int r = 0; r < 8; ++r)
    pre[(Mb + r) * Hd + N] = acc[r] + bias;
}

// ---------------------------------------------------------------------------
// Kernel 2: hidden = tanh(pre + sum_j (w[i,j]+alpha[i,j]*hebb[b,i,j])*prev[b,j])
// Pure streaming (hebb used once): one wave per (b,i) row, b128 loads,
// shuffle-tree reduction. pre is staged in-place in the hidden output slot.
// ---------------------------------------------------------------------------
__global__ __launch_bounds__(256) void k_hidden(const float* __restrict__ hebb,
                                                const float* __restrict__ w,
                                                const float* __restrict__ alpha,
                                                const float* __restrict__ prev,
                                                float* __restrict__ hid) {
  const int warp = threadIdx.x >> 5;
  const int lane = threadIdx.x & 31;
  const int row  = blockIdx.x * 8 + warp;       // row in [0, 64*1024)
  const int b    = row >> 10;
  const int i    = row & 1023;
  const v4f* hb = (const v4f*)(hebb + ((size_t)row << 10));
  const v4f* wr = (const v4f*)(w     + ((size_t)i  << 10));
  const v4f* ar = (const v4f*)(alpha + ((size_t)i  << 10));
  const v4f* pv = (const v4f*)(prev  + (b << 10));
  float s = 0.0f;
#pragma unroll
  for (int it = 0; it < 8; ++it) {
    const int j4 = lane + it * 32;              // 256 float4 per row
    v4f h4 = hb[j4], w4 = wr[j4], a4 = ar[j4], p4 = pv[j4];
    s += (w4.x + a4.x * h4.x) * p4.x;
    s += (w4.y + a4.y * h4.y) * p4.y;
    s += (w4.z + a4.z * h4.z) * p4.z;
    s += (w4.w + a4.w * h4.w) * p4.w;
  }
#pragma unroll
  for (int off = 16; off; off >>= 1) s += __shfl_xor(s, off, 32);
  if (lane == 0) hid[row] = tanhf(hid[row] + s);
}

// ---------------------------------------------------------------------------
// Kernel 3: a_out[64,256] = hidden[64,1024] @ Wh2o^T + bh2o  (fp32 WMMA)
// 64 tiles, one wave each, K-loop of 256 steps.
// ---------------------------------------------------------------------------
__global__ __launch_bounds__(256) void k_aout(const float* __restrict__ hid,
                                              const float* __restrict__ Wh2o,
                                              const float* __restrict__ bh2o,
                                              float* __restrict__ a_out) {
  const int warp = threadIdx.x >> 5;
  const int lane = threadIdx.x & 31;
  const int t    = blockIdx.x * 8 + warp;       // tile id in [0,64)
  const int m0   = (t >> 4) << 4;               // 4 batch tiles
  const int n0   = (t & 15) << 4;               // 16 output tiles
  const int l16  = lane & 15;
  const int half = lane >> 4;
  const float* arow = hid  + (m0 + l16) * Hd + (half << 1);
  const float* brow = Wh2o + (n0 + l16) * Hd + (half << 1);
  v8f acc = {};
  for (int k = 0; k < Hd; k += 4) {
    v2f a = *(const v2f*)(arow + k);
    v2f b = *(const v2f*)(brow + k);
    acc = __builtin_amdgcn_wmma_f32_16x16x4_f32(false, a, false, b,
                                                (short)0, acc, false, false);
  }
  const int Mb = m0 + (half << 3);
  const int N  = n0 + l16;
  const float bias = bh2o[N];
#pragma unroll
  for (int r = 0; r < 8; ++r)
    a_out[(Mb + r) * Od + N] = acc[r] + bias;
}

// ---------------------------------------------------------------------------
// Kernel 4: v_out[b] = hidden[b]·Wh2v + bh2v ;  M_out[b] = tanh(hidden[b]·Wmod + bmod)
// (M == 1, so the mean over the neuromod axis is just tanh of one dot.)
// One wave per batch.
// ---------------------------------------------------------------------------
__global__ __launch_bounds__(64) void k_vmod(const float* __restrict__ hid,
                                             const float* __restrict__ Wh2v,
                                             const float* __restrict__ bh2v,
                                             const float* __restrict__ Wmod,
                                             const float* __restrict__ bmod,
                                             float* __restrict__ v_out,
                                             float* __restrict__ m_out) {
  const int warp = threadIdx.x >> 5;
  const int lane = threadIdx.x & 31;
  const int b = blockIdx.x * 2 + warp;
  const float* h = hid + (b << 10);
  float sv = 0.0f, sm = 0.0f;
  for (int j = lane; j < Hd; j += 32) {
    const float hv = h[j];
    sv += hv * Wh2v[j];
    sm += hv * Wmod[j];
  }
#pragma unroll
  for (int off = 16; off; off >>= 1) {
    sv += __shfl_xor(sv, off, 32);
    sm += __shfl_xor(sm, off, 32);
  }
  if (lane == 0) {
    v_out[b] = sv + bh2v[0];
    m_out[b] = tanhf(sm + bmod[0]);
  }
}

// ---------------------------------------------------------------------------
// Kernel 5: hebb_new = clip(hebb + M_out[b]*hidden[b,i]*prev[b,j], -1, 1)
// 16.7M float4 items; nontemporal b128 stream so 512 MB doesn't thrash L2.
// ---------------------------------------------------------------------------
__global__ __launch_bounds__(256) void k_hebb(const float* __restrict__ hebb,
                                              const float* __restrict__ prev,
                                              const float* __restrict__ hid,
                                              const float* __restrict__ m_out,
                                              float* __restrict__ hebb_out) {
  const size_t idx = (size_t)blockIdx.x * 256 + threadIdx.x; // float4 index
  const int b   = (int)(idx >> 18);          // H*H/4 = 262144 = 2^18
  const int rem = (int)(idx & 262143);
  const int i   = rem >> 8;                  // H/4 = 256
  const int j4  = rem & 255;
  const float scale = m_out[b] * hid[(b << 10) + i];
  const v4f* hb = (const v4f*)hebb;
  const v4f* pv = (const v4f*)(prev + (b << 10));
  v4f h4 = __builtin_nontemporal_load(hb + idx);
  v4f p4 = pv[j4];
  v4f o  = h4 + scale * p4;
  o.x = fminf(1.0f, fmaxf(-1.0f, o.x));
  o.y = fminf(1.0f, fmaxf(-1.0f, o.y));
  o.z = fminf(1.0f, fmaxf(-1.0f, o.z));
  o.w = fminf(1.0f, fmaxf(-1.0f, o.w));
  __builtin_nontemporal_store(o, ((v4f*)hebb_out) + idx);
}

extern "C" void kernel_launch(void* const* d_in, const int* in_sizes, int n_in,
                              void* d_out, int out_size, void* d_ws, size_t ws_size,
                              hipStream_t stream) {
  const float* inputs = (const float*)d_in[0];
  const float* prev   = (const float*)d_in[1];
  const float* hebb   = (const float*)d_in[2];
  const float* w      = (const float*)d_in[3];
  const float* alpha  = (const float*)d_in[4];
  const float* Wi2h   = (const float*)d_in[5];
  const float* bi2h   = (const float*)d_in[6];
  const float* Wh2o   = (const float*)d_in[7];
  const float* bh2o   = (const float*)d_in[8];
  const float* Wh2v   = (const float*)d_in[9];
  const float* bh2v   = (const float*)d_in[10];
  const float* Wmod   = (const float*)d_in[11];
  const float* bmod   = (const float*)d_in[12];

  float* out      = (float*)d_out;
  float* a_out    = out;                       // [64,256]
  float* v_out    = a_out + Bsz * Od;          // [64,1]
  float* hid      = v_out + Bsz;               // [64,1024] (also stages i2h preact)
  float* hebb_out = hid + Bsz * Hd;            // [64,1024,1024]
  float* m_out    = (float*)d_ws;              // 64 floats scratch

  // 1) i2h preactivation (WMMA f32) -> staged into hidden slot of d_out
  k_i2h<<<32, 256, 0, stream>>>(inputs, Wi2h, bi2h, hid);
  // 2) plastic mat-vec + tanh (streams hebb once), overwrites preact with hidden
  k_hidden<<<8192, 256, 0, stream>>>(hebb, w, alpha, prev, hid);
  // 3) actor head (WMMA f32)
  k_aout<<<8, 256, 0, stream>>>(hid, Wh2o, bh2o, a_out);
  // 4) value head + neuromodulatory gate
  k_vmod<<<32, 64, 0, stream>>>(hid, Wh2v, bh2v, Wmod, bmod, v_out, m_out);
  // 5) hebbian outer-product update + clamp (nontemporal stream)
  k_hebb<<<65536, 256, 0, stream>>>(hebb, prev, hid, m_out, hebb_out);
}